// ODEFunc2_interaction_energy_12378095747590
// MI455X (gfx1250) — compile-verified
//
#include <hip/hip_runtime.h>
#include <math.h>

typedef __attribute__((ext_vector_type(2))) float v2f;
typedef __attribute__((ext_vector_type(8))) float v8f;

__device__ __forceinline__ float fast_tanh(float x) {
    float e = __expf(-2.0f * fabsf(x));
    float t = (1.0f - e) * __builtin_amdgcn_rcpf(1.0f + e);
    return copysignf(t, x);
}

__device__ __forceinline__ float fast_sech2(float x) {
    // sech^2 is even: overflow-safe via e = exp(-2|x|) in (0,1]
    float e = __expf(-2.0f * fabsf(x));
    float u = 1.0f + e;
    return 4.0f * e * __builtin_amdgcn_rcpf(u * u);
}

// ---------------- Kernel A: per-particle MLP + pointwise outputs ----------------
__global__ __launch_bounds__(256) void mlp_kernel(
    const float* __restrict__ t, const float* __restrict__ z,
    const float* __restrict__ lnw, const float* __restrict__ W1,
    const float* __restrict__ b1, const float* __restrict__ Wv,
    const float* __restrict__ bv, const float* __restrict__ Wg,
    const float* __restrict__ bg,
    float* __restrict__ vbuf, float* __restrict__ wexp,
    float* __restrict__ out, int n)
{
    int i = blockIdx.x * blockDim.x + threadIdx.x;
    if (i >= n) return;
    float z0 = z[i * 3 + 0], z1 = z[i * 3 + 1], z2 = z[i * 3 + 2];
    float tt = t[0];
    float v0 = bv[0], v1 = bv[1], v2 = bv[2], g = bg[0];
#pragma unroll 4
    for (int k = 0; k < 64; ++k) {
        float pre = fmaf(z0, W1[k],
                    fmaf(z1, W1[64 + k],
                    fmaf(z2, W1[128 + k],
                    fmaf(tt, W1[192 + k], b1[k]))));
        float h = fast_tanh(pre);
        v0 = fmaf(h, Wv[k * 3 + 0], v0);
        v1 = fmaf(h, Wv[k * 3 + 1], v1);
        v2 = fmaf(h, Wv[k * 3 + 2], v2);
        g  = fmaf(h, Wg[k], g);
    }
    float w = __expf(lnw[i]);
    vbuf[i * 3 + 0] = v0;
    vbuf[i * 3 + 1] = v1;
    vbuf[i * 3 + 2] = v2;
    wexp[i] = w;
    out[3 * n + i] = g;                                          // dlnw_dt
    out[4 * n + i] = (v0 * v0 + v1 * v1 + v2 * v2 + g * g) * w;  // dm_dt
}

// ---------------- Kernel B: pairwise force via WMMA Gram tiles ----------------
// One wave per (i-tile of 16, j-chunk). d2_ij = r2_i + r2_j - 2*(z_i . z_j),
// Gram tile computed by V_WMMA_F32_16X16X4_F32 (K=4, 4th component zero-padded).
// The j-tile z loads are software-pipelined (1-deep) so the L2 latency of each
// global_load_b96 overlaps the WMMA + screen math of the previous tile.
__global__ __launch_bounds__(256) void pair_force_kernel(
    const float* __restrict__ z, const float* __restrict__ wexp,
    const float* __restrict__ Wi1, const float* __restrict__ bi1,
    const float* __restrict__ Wi2,
    float* __restrict__ partial, int n, int jchunks, int tilesPerChunk)
{
    const int lane = threadIdx.x & 31;
    const int wave = blockIdx.x * (blockDim.x >> 5) + (threadIdx.x >> 5);
    const int it = wave / jchunks;   // i-tile index
    const int jc = wave % jchunks;   // j-chunk index
    const int hi = lane >> 4;        // 0: lanes 0-15, 1: lanes 16-31
    const int col = lane & 15;
    const int i0 = it * 16;

    // Uniform interaction parameters (scalar-loaded)
    float w1k[8], b1k[8], ck[8];
#pragma unroll
    for (int k = 0; k < 8; ++k) {
        w1k[k] = Wi1[k];
        b1k[k] = bi1[k];
        ck[k]  = Wi1[k] * Wi2[k];
    }

    // A matrix (16x4 f32): lanes 0-15 hold K=0,1 of row M=col; lanes 16-31 hold K=2,3
    const int rI = i0 + col;
    float zr0 = z[rI * 3 + 0], zr1 = z[rI * 3 + 1], zr2 = z[rI * 3 + 2];
    v2f aM;
    aM.x = hi ? zr2 : zr0;
    aM.y = hi ? 0.0f : zr1;
    float r2row = zr0 * zr0 + zr1 * zr1 + zr2 * zr2;

    // Broadcast i-row data so each lane has all 8 rows it covers in the C tile
    float zi0[8], zi1[8], zi2[8], r2i[8];
#pragma unroll
    for (int p = 0; p < 8; ++p) {
        int src = p + hi * 8;
        zi0[p] = __shfl(zr0, src, 32);
        zi1[p] = __shfl(zr1, src, 32);
        zi2[p] = __shfl(zr2, src, 32);
        r2i[p] = __shfl(r2row, src, 32);
    }

    float acc[8][3];
#pragma unroll
    for (int p = 0; p < 8; ++p) {
        acc[p][0] = 0.0f; acc[p][1] = 0.0f; acc[p][2] = 0.0f;
    }

    // Pipeline prologue: load the first j-tile's z for this lane's column
    int j = jc * tilesPerChunk * 16 + col;
    float c0 = z[j * 3 + 0], c1 = z[j * 3 + 1], c2 = z[j * 3 + 2];

    for (int s = 0; s < tilesPerChunk; ++s) {
        // Issue next tile's load now (clamped on the last iteration so the
        // address stays valid without a branch); consumed at loop bottom.
        int jn = j + 16;
        int jl = (jn < n) ? jn : j;
        float n0 = z[(size_t)jl * 3 + 0];
        float n1 = z[(size_t)jl * 3 + 1];
        float n2 = z[(size_t)jl * 3 + 2];

        float r2j = c0 * c0 + c1 * c1 + c2 * c2;

        // B matrix (4x16 f32), mirrored K-split layout
        v2f bM;
        bM.x = hi ? c2 : c0;
        bM.y = hi ? 0.0f : c1;

        v8f cz = {0.0f, 0.0f, 0.0f, 0.0f, 0.0f, 0.0f, 0.0f, 0.0f};
        // Gram tile: G[M][N] = z_{i0+M} . z_{j0+N}
        v8f gm = __builtin_amdgcn_wmma_f32_16x16x4_f32(
            false, aM, false, bM, (short)0, cz, false, false);

        // Distance screen: dist < 0.5  <=>  d2 (incl. EPS) < 0.25.
        float d2a[8];
#pragma unroll
        for (int p = 0; p < 8; ++p) {
            d2a[p] = fmaxf(fmaf(-2.0f, gm[p], r2i[p] + r2j), 0.0f) + 1e-8f;
        }
        float dmin = fminf(fminf(fminf(d2a[0], d2a[1]), fminf(d2a[2], d2a[3])),
                           fminf(fminf(d2a[4], d2a[5]), fminf(d2a[6], d2a[7])));

        // Tile-level vote: uniform scalar branch, EXEC stays all-ones for the
        // next WMMA. Common case (no neighbors in tile) skips everything below.
        if (__any(dmin < 0.25f)) {
            float wj = wexp[j];   // rare path only
#pragma unroll
            for (int p = 0; p < 8; ++p) {
                int irow = i0 + p + hi * 8;
                if (d2a[p] < 0.25f && irow != j) {
                    float rinv = __builtin_amdgcn_rsqf(d2a[p]);  // 1/dist
                    float dist = d2a[p] * rinv;
                    float dU = 0.0f;
#pragma unroll
                    for (int k = 0; k < 8; ++k) {
                        dU = fmaf(ck[k], fast_sech2(fmaf(dist, w1k[k], b1k[k])), dU);
                    }
                    float coeff = -(wj * dU) * rinv * 0.0625f;   // /M_PARTICLES
                    acc[p][0] = fmaf(coeff, zi0[p] - c0, acc[p][0]);
                    acc[p][1] = fmaf(coeff, zi1[p] - c1, acc[p][1]);
                    acc[p][2] = fmaf(coeff, zi2[p] - c2, acc[p][2]);
                }
            }
        }

        // Rotate pipeline registers
        j = jn;
        c0 = n0; c1 = n1; c2 = n2;
    }

    // Reduce over the 16 columns within each half-wave (xor<16 stays in half)
#pragma unroll
    for (int p = 0; p < 8; ++p) {
#pragma unroll
        for (int d = 0; d < 3; ++d) {
            float v = acc[p][d];
            v += __shfl_xor(v, 1, 32);
            v += __shfl_xor(v, 2, 32);
            v += __shfl_xor(v, 4, 32);
            v += __shfl_xor(v, 8, 32);
            acc[p][d] = v;
        }
    }

    // Lane q (q = col < 8) writes row i0 + q + hi*8. Select with unrolled
    // compares (constant indices) to avoid scratch spills.
    if (col < 8) {
        float o0 = 0.0f, o1 = 0.0f, o2 = 0.0f;
#pragma unroll
        for (int p = 0; p < 8; ++p) {
            if (col == p) { o0 = acc[p][0]; o1 = acc[p][1]; o2 = acc[p][2]; }
        }
        int row = i0 + col + hi * 8;
        size_t base = ((size_t)jc * (size_t)n + (size_t)row) * 3;
        partial[base + 0] = o0;
        partial[base + 1] = o1;
        partial[base + 2] = o2;
    }
}

// ---------------- Kernel C: deterministic reduction + add drift v ----------------
__global__ __launch_bounds__(256) void finalize_kernel(
    const float* __restrict__ vbuf, const float* __restrict__ partial,
    float* __restrict__ out, int n, int jchunks)
{
    int idx = blockIdx.x * blockDim.x + threadIdx.x;
    if (idx >= 3 * n) return;
    float s = vbuf[idx];
    for (int c = 0; c < jchunks; ++c) {
        s += partial[(size_t)c * 3 * (size_t)n + idx];
    }
    out[idx] = s;  // dz_dt
}

extern "C" void kernel_launch(void* const* d_in, const int* in_sizes, int n_in,
                              void* d_out, int out_size, void* d_ws, size_t ws_size,
                              hipStream_t stream) {
    const float* t   = (const float*)d_in[0];
    const float* z   = (const float*)d_in[1];
    const float* lnw = (const float*)d_in[2];
    const float* W1  = (const float*)d_in[3];
    const float* b1  = (const float*)d_in[4];
    const float* Wv  = (const float*)d_in[5];
    const float* bv  = (const float*)d_in[6];
    const float* Wg  = (const float*)d_in[7];
    const float* bg  = (const float*)d_in[8];
    const float* Wi1 = (const float*)d_in[9];
    const float* bi1 = (const float*)d_in[10];
    const float* Wi2 = (const float*)d_in[11];
    // d_in[12] (bi2) does not affect dU/dr -> unused

    float* out = (float*)d_out;
    const int n = in_sizes[1] / 3;             // 4096

    float* ws      = (float*)d_ws;
    float* vbuf    = ws;                        // [n*3]
    float* wexp    = ws + (size_t)3 * n;        // [n]
    float* partial = ws + (size_t)4 * n;        // [JS][n][3]

    const int JS = 16;                          // j-chunks per i-tile
    const int tiles = n / 16;                   // 256
    const int tilesPerChunk = tiles / JS;       // 16

    mlp_kernel<<<(n + 255) / 256, 256, 0, stream>>>(
        t, z, lnw, W1, b1, Wv, bv, Wg, bg, vbuf, wexp, out, n);

    const int waves = tiles * JS;               // 4096 waves
    pair_force_kernel<<<waves / 8, 256, 0, stream>>>(
        z, wexp, Wi1, bi1, Wi2, partial, n, JS, tilesPerChunk);

    finalize_kernel<<<(3 * n + 255) / 256, 256, 0, stream>>>(
        vbuf, partial, out, n, JS);
}